// RGCN_Layer_24550033063975
// MI455X (gfx1250) — compile-verified
//
#include <hip/hip_runtime.h>
#include <hip/hip_bf16.h>

typedef __attribute__((ext_vector_type(2))) float v2f;
typedef __attribute__((ext_vector_type(8))) float v8f;

#define N_ENT   100000
#define BATCH   50000
#define DIM     128
#define ODIM    128
#define NREL    8
#define NEDGE   200000

// ---------------------------------------------------------------------------
// Stage 0: zero the [R, N_ENT, D] accumulator in workspace (409.6 MB)
// ---------------------------------------------------------------------------
__global__ void zero_ws_kernel(float4* __restrict__ p, long n4) {
    long i = (long)blockIdx.x * blockDim.x + threadIdx.x;
    long stride = (long)gridDim.x * blockDim.x;
    float4 z; z.x = 0.f; z.y = 0.f; z.z = 0.f; z.w = 0.f;
    for (; i < n4; i += stride) p[i] = z;
}

// ---------------------------------------------------------------------------
// Stage 1: per-relation SpMM scatter-add.
// One wave32 per edge; lane l owns channels [4l, 4l+4).
// ws[r, src, :] += val * embeddings[dst, :]
// ---------------------------------------------------------------------------
__global__ __launch_bounds__(256) void spmm_kernel(
    const float* __restrict__ emb,
    const float* __restrict__ edge_val,
    const int*   __restrict__ edge_src,
    const int*   __restrict__ edge_dst,
    float*       __restrict__ ws)
{
    int wid  = (blockIdx.x * blockDim.x + threadIdx.x) >> 5;   // edge id (r*E+e)
    int lane = threadIdx.x & 31;
    if (wid >= NREL * NEDGE) return;                            // wave-uniform

    float v = edge_val[wid];
    int   s = edge_src[wid];
    int   d = edge_dst[wid];
    int   r = wid / NEDGE;

    const float4 x = *(const float4*)(emb + (size_t)d * DIM + lane * 4);
    float* o = ws + ((size_t)r * N_ENT + (size_t)s) * DIM + lane * 4;
    unsafeAtomicAdd(o + 0, v * x.x);
    unsafeAtomicAdd(o + 1, v * x.y);
    unsafeAtomicAdd(o + 2, v * x.z);
    unsafeAtomicAdd(o + 3, v * x.w);
}

// ---------------------------------------------------------------------------
// Stage 2: gathered GEMM + sigmoid on the WMMA pipe (f32, 16x16x4).
// Block = 320 threads (10 waves); each wave owns a 16-row x 128-col output
// tile (head or tail). The 9 shared B matrices (self_kernel + 8 relation
// kernels) are staged segment-by-segment into LDS in a pair-interleaved
// layout so the hot loop reads (B[k][n], B[k+1][n]) as one ds_load_b64.
//
// WMMA f32 16x16x4 layouts (ISA 7.12.2):
//   A 16x4:  lanes 0-15 -> K = {k0, k0+1}; lanes 16-31 -> {k0+2, k0+3}; M = lane&15
//   C/D:     VGPR v, lanes 0-15 -> M=v, lanes 16-31 -> M=v+8; N = lane&15
// ---------------------------------------------------------------------------
#define WAVES_PER_BLOCK 10
#define TPB (WAVES_PER_BLOCK * 32)

__global__ __launch_bounds__(TPB) void rgcn_wmma_kernel(
    const float* __restrict__ head_e,
    const float* __restrict__ tail_e,
    const float* __restrict__ self_k,   // [D, O]
    const float* __restrict__ rel_k,    // [R, D, O]
    const int*   __restrict__ head_idx,
    const int*   __restrict__ tail_idx,
    const float* __restrict__ ws,       // [R, N_ENT, D]
    float*       __restrict__ out)      // [2, BATCH, O]
{
    __shared__ float lbs[DIM * ODIM];   // 64 KB: one B segment, pair-interleaved

    const int tid  = threadIdx.x;
    const int w    = blockIdx.x * WAVES_PER_BLOCK + (tid >> 5);
    const int lane = tid & 31;
    const int lo   = lane & 15;
    const int hi   = lane >> 4;                  // 0 or 1
    const int koff = hi * 2;

    const int which    = w & 1;                  // 0 = head, 1 = tail
    const int row_tile = w >> 1;
    const int row      = row_tile * 16 + lo;     // M = lo for both half-waves

    const float* inp = which ? tail_e : head_e;
    const int    idx = (which ? tail_idx : head_idx)[row];

    v8f acc[8];
#pragma unroll
    for (int i = 0; i < 8; ++i) acc[i] = (v8f)0.0f;

    for (int seg = 0; seg < 1 + NREL; ++seg) {
        const float* Bm = (seg == 0) ? self_k : (rel_k + (size_t)(seg - 1) * DIM * ODIM);

        // ---- cooperative stage of B into LDS (pair-interleaved) ----
        __syncthreads();   // previous segment's LDS reads are done
        for (int i = tid * 4; i < DIM * ODIM; i += TPB * 4) {
            const float4 v4 = *(const float4*)(Bm + i);
            const int k = i >> 7;                // row (K index)
            const int n = i & 127;               // col (N index)
            const int base = ((k >> 1) << 8) + 2 * n + (k & 1);
            lbs[base + 0] = v4.x;
            lbs[base + 2] = v4.y;
            lbs[base + 4] = v4.z;
            lbs[base + 6] = v4.w;
        }
        __syncthreads();

        // ---- A row for this segment ----
        const float* Arow = (seg == 0)
            ? (inp + (size_t)row * DIM)
            : (ws + ((size_t)(seg - 1) * N_ENT + (size_t)idx) * DIM);

        // ---- K loop: 1 global b64 (A) + 8 ds b64 (B) + 8 WMMA per step ----
        for (int k0 = 0; k0 < DIM; k0 += 4) {
            const int k = k0 + koff;             // even
            v2f a;
            a.x = Arow[k];
            a.y = Arow[k + 1];
            const int brow = (k >> 1) << 8;
#pragma unroll
            for (int nt = 0; nt < 8; ++nt) {
                const v2f b = *(const v2f*)(&lbs[brow + 2 * (nt * 16 + lo)]);
                acc[nt] = __builtin_amdgcn_wmma_f32_16x16x4_f32(
                    false, a, false, b, (short)0, acc[nt], false, false);
            }
        }
    }

    // ---- sigmoid + store ----
    float* obase = out + (size_t)which * BATCH * ODIM;
#pragma unroll
    for (int nt = 0; nt < 8; ++nt) {
#pragma unroll
        for (int v = 0; v < 8; ++v) {
            const int m  = v + hi * 8;
            const float x  = acc[nt][v];
            const float sg = 1.0f / (1.0f + __expf(-x));
            obase[(size_t)(row_tile * 16 + m) * ODIM + nt * 16 + lo] = sg;
        }
    }
}

// ---------------------------------------------------------------------------
extern "C" void kernel_launch(void* const* d_in, const int* in_sizes, int n_in,
                              void* d_out, int out_size, void* d_ws, size_t ws_size,
                              hipStream_t stream) {
    const float* emb    = (const float*)d_in[0];
    const float* head_e = (const float*)d_in[1];
    const float* tail_e = (const float*)d_in[2];
    const float* rel_k  = (const float*)d_in[3];
    const float* self_k = (const float*)d_in[4];
    const float* e_val  = (const float*)d_in[5];
    const int*   h_idx  = (const int*)d_in[6];
    const int*   t_idx  = (const int*)d_in[7];
    const int*   e_src  = (const int*)d_in[8];
    const int*   e_dst  = (const int*)d_in[9];
    float* out = (float*)d_out;
    float* ws  = (float*)d_ws;

    const long WS_FLOATS = (long)NREL * N_ENT * DIM;   // 102.4M floats

    // Stage 0: clear accumulator
    zero_ws_kernel<<<4096, 256, 0, stream>>>((float4*)ws, WS_FLOATS / 4);

    // Stage 1: scatter-add SpMM (1 wave per edge, 8 waves per block)
    const int totalEdges = NREL * NEDGE;               // 1.6M
    const int spmmBlocks = (totalEdges + 7) / 8;
    spmm_kernel<<<spmmBlocks, 256, 0, stream>>>(emb, e_val, e_src, e_dst, ws);

    // Stage 2: WMMA gathered GEMM + sigmoid.
    // 6250 wave-work-items / 10 waves per block = 625 blocks exactly.
    const int numWaves   = (BATCH / 16) * 2;           // 6250
    const int gemmBlocks = numWaves / WAVES_PER_BLOCK; // 625
    rgcn_wmma_kernel<<<gemmBlocks, TPB, 0, stream>>>(
        head_e, tail_e, self_k, rel_k, h_idx, t_idx, ws, out);
}